// CRNN_4655744549062
// MI455X (gfx1250) — compile-verified
//
#include <hip/hip_runtime.h>
#include <hip/hip_bf16.h>

typedef __attribute__((ext_vector_type(2))) float v2f;
typedef __attribute__((ext_vector_type(4))) float v4f;
typedef __attribute__((ext_vector_type(8))) float v8f;

#define RNN_SIZE   512
#define NTR        64
#define TSTEPS     1000
#define DT_C       0.1f
#define SQRT_DT_C  0.316227766016837933f

#define NWG        16
#define THREADS    256
#define JROWS      (RNN_SIZE / NWG)   // 32 neuron rows per workgroup
#define J_STRIDE   516                 // padded LDS row stride (conflict-free b64 reads)
#define TH_STRIDE  516

__device__ __forceinline__ float dtanh(float x) {
#if __has_builtin(__builtin_amdgcn_tanhf)
    return __builtin_amdgcn_tanhf(x);   // v_tanh_f32 (gfx1250 TRANS op)
#else
    return tanhf(x);
#endif
}

__device__ __forceinline__ void wait_asynccnt0() {
#if __has_builtin(__builtin_amdgcn_s_wait_asynccnt)
    __builtin_amdgcn_s_wait_asynccnt(0);
#else
    asm volatile("s_wait_asynccnt 0" ::: "memory");
#endif
}

// Async global -> LDS copy of 16 bytes per lane (CDNA5 GLOBAL_LOAD_ASYNC_TO_LDS_B128,
// GV addressing mode; tracked by ASYNCcnt).
__device__ __forceinline__ void async_g2lds_b128(unsigned lds_byte_addr, const void* gaddr) {
    asm volatile("global_load_async_to_lds_b128 %0, %1, off"
                 :: "v"(lds_byte_addr), "v"((unsigned long long)(uintptr_t)gaddr)
                 : "memory");
}

// Software grid barrier across the 16 persistent workgroups.
// Monotonic counter; init kernel zeroes it each launch (ws is poisoned 0xAA once).
__device__ __forceinline__ void grid_sync(unsigned* cnt, unsigned target) {
    __threadfence();                    // make this thread's global stores agent-visible
    __syncthreads();
    if (threadIdx.x == 0) {
        __hip_atomic_fetch_add(cnt, 1u, __ATOMIC_RELEASE, __HIP_MEMORY_SCOPE_AGENT);
        while (__hip_atomic_load(cnt, __ATOMIC_ACQUIRE, __HIP_MEMORY_SCOPE_AGENT) < target) {
            __builtin_amdgcn_s_sleep(1);
        }
    }
    __syncthreads();
    __builtin_amdgcn_fence(__ATOMIC_ACQUIRE, "agent");  // invalidate caches: see peers' writes
}

__global__ void crnn_init_kernel(const float* __restrict__ s0,
                                 float* __restrict__ xseq,
                                 float* __restrict__ rseq,
                                 float* __restrict__ outseq,
                                 unsigned* __restrict__ cnt) {
    const int tid    = blockIdx.x * blockDim.x + threadIdx.x;
    const int stride = gridDim.x * blockDim.x;
    if (tid == 0) *cnt = 0u;
    for (int i = tid; i < NTR * RNN_SIZE; i += stride) {
        const int n = i / RNN_SIZE, j = i % RNN_SIZE;
        const size_t base = (size_t)n * (TSTEPS + 1) * RNN_SIZE;
        xseq[base + j] = s0[j];                                   // x_seq[:,0,:]
        rseq[base + (size_t)TSTEPS * RNN_SIZE + j] = 0.0f;        // r_seq[:,T,:] = 0
    }
    for (int i = tid; i < NTR * TSTEPS * 2; i += stride) outseq[i] = 0.0f;  // for atomics
}

__global__ void __launch_bounds__(THREADS, 1)
crnn_scan_kernel(const float* __restrict__ inp,    // [64][1000][2]
                 const float* __restrict__ s0,     // [512]
                 const float* __restrict__ Bm,     // [512][2]
                 const float* __restrict__ J,      // [512][512]
                 const float* __restrict__ W,      // [2][512]
                 const float* __restrict__ sig,    // [512]
                 const float* __restrict__ eps,    // [64][1000][512]
                 const float* __restrict__ samps,  // [64][1000][512]
                 float* __restrict__ xseq,
                 float* __restrict__ rseq,
                 float* __restrict__ outseq,
                 unsigned* __restrict__ cnt,
                 float* __restrict__ th0,          // [64][512] ping
                 float* __restrict__ th1)          // [64][512] pong
{
    extern __shared__ float smem[];
    float* sm_j  = smem;                           // [32][516] resident J slice
    float* sm_th = smem + JROWS * J_STRIDE;        // [64][516] staged tanh(x)

    const int tid    = threadIdx.x;
    const int j_base = blockIdx.x * JROWS;

    // ---- one-time: J slice (rows j_base..j_base+31) -> LDS ----
    for (int f = tid; f < JROWS * (RNN_SIZE / 4); f += THREADS) {
        const int row = f / (RNN_SIZE / 4);
        const int c4  = f % (RNN_SIZE / 4);
        v4f val = *(const v4f*)(J + (size_t)(j_base + row) * RNN_SIZE + c4 * 4);
        *(v4f*)(sm_j + row * J_STRIDE + c4 * 4) = val;
    }
    // ---- one-time: th0[n][j] = tanh(x0[j]) for our j slice, all trials ----
    for (int i = tid; i < NTR * JROWS; i += THREADS) {
        const int n = i / JROWS;
        const int j = j_base + (i % JROWS);
        th0[n * RNN_SIZE + j] = dtanh(s0[j]);
    }

    const int lane   = tid & 31;
    const int wave   = tid >> 5;      // 8 waves
    const int m_tile = wave & 3;      // trial tile (4 x 16 trials)
    const int n_tile = wave >> 2;     // neuron tile within slice (2 x 16)
    const int half   = lane >> 4;     // K-half selector per WMMA layout
    const int l16    = lane & 15;
    const int jrl    = n_tile * 16 + l16;
    const int j      = j_base + jrl;  // this lane's neuron column (constant)

    const float Bj0   = Bm[j * 2 + 0];
    const float Bj1   = Bm[j * 2 + 1];
    const float w0    = W[j];
    const float w1    = W[RNN_SIZE + j];
    const float sgv   = sig[j];
    const float sig2j = sgv * sgv;
    const float x0j   = s0[j];

    float xv[8];
    #pragma unroll
    for (int v = 0; v < 8; ++v) xv[v] = x0j;      // x state lives in registers

    const float* jB  = sm_j + jrl * J_STRIDE + 2 * half;
    const float* thA = sm_th + (m_tile * 16 + l16) * TH_STRIDE + 2 * half;
    const unsigned sm_th_base = (unsigned)(uintptr_t)sm_th;   // LDS byte offset of sm_th

    unsigned bar = (unsigned)gridDim.x;
    grid_sync(cnt, bar);   // all th0 slices + LDS J ready

    for (int t = 0; t < TSTEPS; ++t) {
        const float* th_rd = (t & 1) ? th1 : th0;
        float*       th_wr = (t & 1) ? th0 : th1;

        // ---- stage full tanh(x) [64][512] -> padded LDS via async-to-LDS DMA ----
        for (int f = tid; f < NTR * (RNN_SIZE / 4); f += THREADS) {
            const int row = f >> 7;        // /128
            const int c4  = f & 127;
            async_g2lds_b128(sm_th_base + (unsigned)(row * TH_STRIDE + c4 * 4) * 4u,
                             th_rd + row * RNN_SIZE + c4 * 4);
        }
        wait_asynccnt0();
        __syncthreads();

        // ---- fp32 WMMA GEMM: D[16x16] += A[16x4] x B[4x16] over K=512 ----
        v8f acc0 = {0.f,0.f,0.f,0.f,0.f,0.f,0.f,0.f};
        v8f acc1 = {0.f,0.f,0.f,0.f,0.f,0.f,0.f,0.f};
        #pragma unroll 4
        for (int k = 0; k < RNN_SIZE; k += 8) {
            v2f a0 = *(const v2f*)(thA + k);
            v2f b0 = *(const v2f*)(jB + k);
            acc0 = __builtin_amdgcn_wmma_f32_16x16x4_f32(
                       false, a0, false, b0, (short)0, acc0, false, false);
            v2f a1 = *(const v2f*)(thA + k + 4);
            v2f b1 = *(const v2f*)(jB + k + 4);
            acc1 = __builtin_amdgcn_wmma_f32_16x16x4_f32(
                       false, a1, false, b1, (short)0, acc1, false, false);
        }
        v8f acc = acc0 + acc1;

        // ---- elementwise update + outputs; D layout: reg v -> trial m_tile*16+v+8*half ----
        #pragma unroll
        for (int v = 0; v < 8; ++v) {
            const int    n  = m_tile * 16 + v + 8 * half;
            const size_t nt = (size_t)n * TSTEPS + t;
            const float  u0 = inp[nt * 2 + 0];
            const float  u1 = inp[nt * 2 + 1];
            const size_t ei = nt * RNN_SIZE + j;
            const float  e  = eps[ei];
            const float  s  = samps[ei];

            const float xn = (1.0f - DT_C) * xv[v]
                           + DT_C * (acc[v] + u0 * Bj0 + u1 * Bj1)
                           + SQRT_DT_C * (e * sig2j) + s;
            xv[v] = xn;
            const float r = dtanh(xn);

            const size_t xb = (size_t)n * (TSTEPS + 1) * RNN_SIZE;
            xseq[xb + (size_t)(t + 1) * RNN_SIZE + j] = xn;
            rseq[xb + (size_t)t * RNN_SIZE + j]       = r;
            th_wr[n * RNN_SIZE + j]                   = r;

            // r @ W.T : reduce 16 j's within each half-wave, then fp32 atomic
            float p0 = r * w0, p1 = r * w1;
            #pragma unroll
            for (int mk = 1; mk < 16; mk <<= 1) {
                p0 += __shfl_xor(p0, mk, 32);
                p1 += __shfl_xor(p1, mk, 32);
            }
            if (l16 == 0) {
                atomicAdd(&outseq[nt * 2 + 0], p0);
                atomicAdd(&outseq[nt * 2 + 1], p1);
            }
        }

        bar += (unsigned)gridDim.x;
        grid_sync(cnt, bar);   // separates th_wr(t) from th_rd(t+1); ping-pong covers the rest
    }
}

extern "C" void kernel_launch(void* const* d_in, const int* in_sizes, int n_in,
                              void* d_out, int out_size, void* d_ws, size_t ws_size,
                              hipStream_t stream) {
    const float* inp  = (const float*)d_in[0];
    const float* s0   = (const float*)d_in[1];
    const float* Bm   = (const float*)d_in[2];
    const float* J    = (const float*)d_in[3];
    const float* W    = (const float*)d_in[4];
    const float* sig  = (const float*)d_in[5];
    const float* eps  = (const float*)d_in[6];
    const float* samp = (const float*)d_in[7];

    float* xseq = (float*)d_out;
    float* rseq = xseq + (size_t)NTR * (TSTEPS + 1) * RNN_SIZE;
    float* outs = rseq + (size_t)NTR * (TSTEPS + 1) * RNN_SIZE;

    unsigned* cnt = (unsigned*)d_ws;
    float* th0 = (float*)((char*)d_ws + 256);
    float* th1 = th0 + NTR * RNN_SIZE;

    crnn_init_kernel<<<256, 256, 0, stream>>>(s0, xseq, rseq, outs, cnt);

    const size_t shmem = (size_t)(JROWS * J_STRIDE + NTR * TH_STRIDE) * sizeof(float);
    crnn_scan_kernel<<<NWG, THREADS, shmem, stream>>>(
        inp, s0, Bm, J, W, sig, eps, samp, xseq, rseq, outs, cnt, th0, th1);
}